// KeyValueMemoryModule_37125697307438
// MI455X (gfx1250) — compile-verified
//
#include <hip/hip_runtime.h>
#include <math.h>

typedef __attribute__((ext_vector_type(2))) float v2f;
typedef __attribute__((ext_vector_type(4))) float v4f;
typedef __attribute__((ext_vector_type(8))) float v8f;

constexpr int B_   = 256;
constexpr int N_   = 4096;
constexpr int KD_  = 64;         // KD == VD == 64
constexpr int ROWQ = KD_ / 4;    // float4 per row = 16
constexpr int THREADS = 256;     // 8 waves (wave32)

__device__ __forceinline__ float sigmoidf_(float x) { return 1.0f / (1.0f + expf(-x)); }

// ---------------------------------------------------------------------------
// Streaming concat copy: out[b, 0..N-1, :] = mem[b, :, :]; out[b, N, :] = tail[b, :]
// float4 granularity, non-temporal (arrays >> 192MB L2, keep it clean).
// ---------------------------------------------------------------------------
__global__ void __launch_bounds__(THREADS) concat_copy_kernel(
    const v4f* __restrict__ mem, const v4f* __restrict__ tail,
    v4f* __restrict__ out, int total)
{
    int idx = blockIdx.x * blockDim.x + threadIdx.x;
    if (idx >= total) return;
    const int perB = (N_ + 1) * ROWQ;
    int b   = idx / perB;
    int rem = idx - b * perB;
    int n   = rem >> 4;          // ROWQ == 16
    int q   = rem & 15;
    v4f v;
    if (n < N_) v = __builtin_nontemporal_load(&mem[((size_t)b * N_ + n) * ROWQ + q]);
    else        v = __builtin_nontemporal_load(&tail[(size_t)b * ROWQ + q]);
    __builtin_nontemporal_store(v, &out[idx]);
}

// ---------------------------------------------------------------------------
// Fused read kernel: one workgroup per batch b.
//   Phase 1: sim[n] = (n<=iter) * <VM[b,n,:], nv[b,:]>  via v_wmma_f32_16x16x4_f32
//   Phase 2: softmax over LDS, conf channel, weighted sum of key_memory rows.
// ---------------------------------------------------------------------------
__global__ void __launch_bounds__(THREADS) kvread_kernel(
    const float* __restrict__ new_value,
    const float* __restrict__ key_memory,
    const float* __restrict__ value_memory,
    const float* __restrict__ gate,
    const float* __restrict__ w_conf,
    const float* __restrict__ b_conf,
    const int*  __restrict__ iteration,
    float* __restrict__ read_out)
{
    __shared__ float s_nv[KD_];
    __shared__ float s_sim[N_];
    __shared__ float s_w[N_];
    __shared__ float s_red[THREADS];
    __shared__ float s_part[4 * KD_];

    const int b    = blockIdx.x;
    const int tid  = threadIdx.x;
    const int lane = tid & 31;
    const int wave = tid >> 5;
    const int iter = iteration[b];

    if (tid < KD_) s_nv[tid] = new_value[(size_t)b * KD_ + tid];
    __syncthreads();

    // ---- Phase 1: masked mat-vec, 16 rows per WMMA tile ------------------
    // A (16x4 f32): lanes 0-15 -> K=0,1 ; lanes 16-31 -> K=2,3 (2 VGPRs)
    // B (4x16 f32): same K striping, value broadcast across all 16 columns
    const float* vm_b = value_memory + (size_t)b * N_ * KD_;
    const int m    = lane & 15;
    const int ksel = (lane >> 4) << 1;   // 0 or 2

    for (int t = wave; t < N_ / 16; t += 8) {          // 32 tiles per wave
        const int n0 = t * 16;
        const float* arow = vm_b + (size_t)(n0 + m) * KD_;
        v8f c = {0.f, 0.f, 0.f, 0.f, 0.f, 0.f, 0.f, 0.f};
        #pragma unroll
        for (int kc = 0; kc < KD_ / 4; ++kc) {         // 16 K-steps of 4
            v2f a = *(const v2f*)(arow + kc * 4 + ksel);
            v2f bb;
            bb.x = s_nv[kc * 4 + ksel];
            bb.y = s_nv[kc * 4 + ksel + 1];
            c = __builtin_amdgcn_wmma_f32_16x16x4_f32(
                    false, a, false, bb, (short)0, c, false, false);
        }
        // D: VGPR r holds M=r (lanes 0-15) / M=8+r (lanes 16-31); all 16
        // columns identical -> lanes 0 and 16 scatter rows, apply mask here.
        if ((lane & 15) == 0) {
            const int nb = n0 + ((lane >> 4) << 3);
            #pragma unroll
            for (int r = 0; r < 8; ++r) {
                const int n = nb + r;
                s_sim[n] = (n <= iter) ? c[r] : 0.0f;
            }
        }
    }
    __syncthreads();

    // ---- softmax: max ----------------------------------------------------
    float lmax = -INFINITY;
    for (int i = tid; i < N_; i += THREADS) lmax = fmaxf(lmax, s_sim[i]);
    s_red[tid] = lmax; __syncthreads();
    for (int s = THREADS / 2; s > 0; s >>= 1) {
        if (tid < s) s_red[tid] = fmaxf(s_red[tid], s_red[tid + s]);
        __syncthreads();
    }
    const float gmax = s_red[0];
    __syncthreads();

    // ---- softmax: exp + sum ---------------------------------------------
    float lsum = 0.f;
    for (int i = tid; i < N_; i += THREADS) {
        float e = expf(s_sim[i] - gmax);
        s_w[i] = e;
        lsum += e;
    }
    s_red[tid] = lsum; __syncthreads();
    for (int s = THREADS / 2; s > 0; s >>= 1) {
        if (tid < s) s_red[tid] += s_red[tid + s];
        __syncthreads();
    }
    const float inv = 1.0f / s_red[0];
    __syncthreads();

    // ---- conf channel: sum_n w[n] * sigmoid(sim[n]*wc + bc) --------------
    const float wc = w_conf[0], bc = b_conf[0];
    float lconf = 0.f;
    for (int i = tid; i < N_; i += THREADS)
        lconf += s_w[i] * sigmoidf_(s_sim[i] * wc + bc);
    s_red[tid] = lconf; __syncthreads();
    for (int s = THREADS / 2; s > 0; s >>= 1) {
        if (tid < s) s_red[tid] += s_red[tid + s];
        __syncthreads();
    }
    const float confsum = s_red[0];
    __syncthreads();

    // ---- Phase 2: read[d] = inv * sum_n w[n] * KM[b,n,d] -----------------
    const int d = tid & (KD_ - 1);
    const int g = tid >> 6;                    // 4 row-groups
    const float* km_b = key_memory + (size_t)b * N_ * KD_;
    float acc = 0.f;
    for (int n = g; n < N_; n += 4)
        acc += s_w[n] * km_b[(size_t)n * KD_ + d];
    s_part[g * KD_ + d] = acc;
    __syncthreads();

    const bool not_first = iter > 1;
    if (g == 0) {
        float tot = (s_part[d] + s_part[KD_ + d] +
                     s_part[2 * KD_ + d] + s_part[3 * KD_ + d]) * inv;
        float r = not_first ? sigmoidf_(gate[(size_t)b * (KD_ + 1) + d]) * tot : 0.0f;
        read_out[(size_t)b * (KD_ + 1) + d] = r;
        if (d == 0) {
            float rc = not_first
                ? sigmoidf_(gate[(size_t)b * (KD_ + 1) + KD_]) * (confsum * inv)
                : 0.0f;
            read_out[(size_t)b * (KD_ + 1) + KD_] = rc;
        }
    }
}

// ---------------------------------------------------------------------------
extern "C" void kernel_launch(void* const* d_in, const int* in_sizes, int n_in,
                              void* d_out, int out_size, void* d_ws, size_t ws_size,
                              hipStream_t stream) {
    const float* new_key      = (const float*)d_in[0];
    const float* new_value    = (const float*)d_in[1];
    const float* key_memory   = (const float*)d_in[2];
    const float* value_memory = (const float*)d_in[3];
    const float* gate         = (const float*)d_in[4];
    const float* w_conf       = (const float*)d_in[5];
    const float* b_conf       = (const float*)d_in[6];
    const int*   iteration    = (const int*)d_in[7];

    float* out_km   = (float*)d_out;
    float* out_vm   = out_km + (size_t)B_ * (N_ + 1) * KD_;
    float* out_read = out_vm + (size_t)B_ * (N_ + 1) * KD_;

    const int total4 = B_ * (N_ + 1) * ROWQ;
    const int blocks = (total4 + THREADS - 1) / THREADS;

    concat_copy_kernel<<<blocks, THREADS, 0, stream>>>(
        (const v4f*)key_memory, (const v4f*)new_key, (v4f*)out_km, total4);
    concat_copy_kernel<<<blocks, THREADS, 0, stream>>>(
        (const v4f*)value_memory, (const v4f*)new_value, (v4f*)out_vm, total4);
    kvread_kernel<<<B_, THREADS, 0, stream>>>(
        new_value, key_memory, value_memory, gate, w_conf, b_conf, iteration, out_read);
}